// SE3PointKernel_3710851744031
// MI455X (gfx1250) — compile-verified
//
#include <hip/hip_runtime.h>
#include <hip/hip_bf16.h>

typedef float v2f __attribute__((ext_vector_type(2)));
typedef float v8f __attribute__((ext_vector_type(8)));

#define NPAIR 24576   // 192*128 point pairs
#define NRC   4096    // 64*64 output (row,col) entries
#define KDIM  36      // 4 radii * 9 SH components

// ---------------------------------------------------------------------------
// Kernel 1: Phi[f][pair] = window_rad(r) * Y_s(u),  f = rad*9 + s
//   s=0: Y00 ; s=1..3: Y1 in reference order [y,z,x] ; s=4..8: Y2 components
// ---------------------------------------------------------------------------
__global__ __launch_bounds__(256) void se3_build_phi(
    const float* __restrict__ diff, float* __restrict__ phi) {
  int idx = blockIdx.x * blockDim.x + threadIdx.x;
  if (idx >= NPAIR) return;
  float dx = diff[3*idx+0], dy = diff[3*idx+1], dz = diff[3*idx+2];
  float r   = sqrtf(dx*dx + dy*dy + dz*dz);
  float inv = 1.0f / (r + 1e-12f);
  float x = dx*inv, y = dy*inv, z = dz*inv;

  float sh[9];
  sh[0] = 0.28209479177387814f;
  const float c1 = 0.4886025119029199f;
  sh[1] = c1*y; sh[2] = c1*z; sh[3] = c1*x;
  const float c2 = 1.0925484305920792f;
  sh[4] = c2*x*y;
  sh[5] = c2*y*z;
  sh[6] = 0.31539156525252005f*(3.0f*z*z - 1.0f);
  sh[7] = c2*x*z;
  sh[8] = 0.5462742152960396f*(x*x - y*y);

  const float radii[4] = {0.0f, 0.8f, 1.6f, 2.4f};
  const float inv_sig  = 1.0f/0.6f;
  const float gnorm    = 1.0f/(2.5066282746310002f*0.6f);
#pragma unroll
  for (int rad = 0; rad < 4; ++rad) {
    float t   = (r - radii[rad])*inv_sig;
    float win = expf(-0.5f*t*t)*gnorm;
#pragma unroll
    for (int s = 0; s < 9; ++s)
      phi[(rad*9+s)*NPAIR + idx] = win*sh[s];
  }
}

// ---------------------------------------------------------------------------
// Kernel 2: T[rc][f]: weights * FVAR folded through the analytic intertwiners.
//   Blocks: A(l0->l0) w[0:1024), B(l1->l0) w[1024:2048),
//           C(l0->l1) w[2048:3072), D(l1->l1) w[3072:6144) (3 J's per (p,q,r))
// ---------------------------------------------------------------------------
__device__ const float Q2TAB[9][5] = {  // (a*3+b) -> l=2 comps, unit Frobenius
  {0.f, 0.f, -0.18257418583505536f, 0.f, -0.31622776601683794f}, // (y,y)
  {0.f, 0.31622776601683794f, 0.f, 0.f, 0.f},                    // (y,z)
  {0.31622776601683794f, 0.f, 0.f, 0.f, 0.f},                    // (y,x)
  {0.f, 0.31622776601683794f, 0.f, 0.f, 0.f},                    // (z,y)
  {0.f, 0.f,  0.36514837167011072f, 0.f, 0.f},                   // (z,z)
  {0.f, 0.f, 0.f, 0.31622776601683794f, 0.f},                    // (z,x)
  {0.31622776601683794f, 0.f, 0.f, 0.f, 0.f},                    // (x,y)
  {0.f, 0.f, 0.f, 0.31622776601683794f, 0.f},                    // (x,z)
  {0.f, 0.f, -0.18257418583505536f, 0.f,  0.31622776601683794f}, // (x,x)
};

__global__ __launch_bounds__(256) void se3_build_T(
    const float* __restrict__ w, float* __restrict__ T) {
  int rc = blockIdx.x * blockDim.x + threadIdx.x;
  if (rc >= NRC) return;
  int row = rc >> 6, col = rc & 63;

  float t[KDIM];
#pragma unroll
  for (int i = 0; i < KDIM; ++i) t[i] = 0.f;

  const float FA = 0.0625f, FB = 0.0625f;
  const float FC = 0.10825317547305482f, FD = 0.0625f;
  const float S3 = 0.57735026918962576f;   // 1/sqrt(3)
  const float S6 = 0.40824829046386302f;   // 1/sqrt(6)

  if (row < 16 && col < 16) {                       // block A: scalar->scalar
    int p = row, q = col;
    for (int rad = 0; rad < 4; ++rad)
      t[rad*9+0] = w[(p*16+q)*4 + rad] * FA;        // J=0 with Y00 in Phi
  } else if (row < 16) {                            // block B: vec->scalar
    int p = row, q = (col-16)/3, b = (col-16)%3;
    for (int rad = 0; rad < 4; ++rad)
      t[rad*9+1+b] = w[1024 + (p*16+q)*4 + rad] * FB * S3;
  } else if (col < 16) {                            // block C: scalar->vec
    int p = (row-16)/3, a = (row-16)%3, q = col;
    for (int rad = 0; rad < 4; ++rad)
      t[rad*9+1+a] = w[2048 + (p*16+q)*4 + rad] * FC * S3;
  } else {                                          // block D: vec->vec, J=0,1,2
    int p = (row-16)/3, a = (row-16)%3;
    int q = (col-16)/3, b = (col-16)%3;
    const float* q2 = Q2TAB[a*3 + b];
    for (int rad = 0; rad < 4; ++rad) {
      const float* wd = w + 3072 + ((p*16+q)*4 + rad)*3;
      float w0 = wd[0]*FD, w1 = wd[1]*FD, w2 = wd[2]*FD;
      if (a == b) t[rad*9+0] += w0 * S3;            // J=0: delta/sqrt3
      if (a != b) {                                 // J=1: Levi-Civita/sqrt6
        int n = 3 - a - b;
        float sgn = (((b - a + 3) % 3) == 1) ? 1.f : -1.f;
        t[rad*9+1+n] += w1 * S6 * sgn;
      }
#pragma unroll
      for (int n = 0; n < 5; ++n)                   // J=2: sym-traceless map
        t[rad*9+4+n] += w2 * q2[n];
    }
  }
#pragma unroll
  for (int i = 0; i < KDIM; ++i) T[rc*KDIM + i] = t[i];
}

// ---------------------------------------------------------------------------
// Kernel 3: OUT(4096 x 24576) = T(4096 x 36) @ Phi(36 x 24576), f32 WMMA.
// One wave = 16 rows x 64 pairs (4 accumulators), 9 K-steps of 16x16x4.
// Grid divides exactly -> EXEC all-ones (WMMA requirement).
// ---------------------------------------------------------------------------
__global__ __launch_bounds__(256) void se3_gemm_wmma(
    const float* __restrict__ T, const float* __restrict__ phi,
    float* __restrict__ out) {
  const int NT = NPAIR;
  int gtid = blockIdx.x * 256 + threadIdx.x;
  int wave = gtid >> 5;
  int lane = threadIdx.x & 31;
  int mt = wave / 384;            // 0..255  (row tile)
  int nt = wave % 384;            // 0..383  (pair tile of 64)
  int rc0 = mt * 16;
  int p0  = nt * 64;
  int half = lane >> 4;           // A/B lane-half selects K pair {0,1}/{2,3}
  int l16  = lane & 15;

  v8f c0 = {}, c1 = {}, c2 = {}, c3 = {};
  const float* arow = T + (size_t)(rc0 + l16)*KDIM + 2*half;

#pragma unroll
  for (int s = 0; s < 9; ++s) {
    v2f a = *(const v2f*)(arow + 4*s);              // A: 16x4 f32 fragment
    int k0 = 4*s + 2*half;
    const float* bp = phi + (size_t)k0*NT + p0 + l16;
    v2f b0, b1, b2, b3;                             // B: 4x16 f32 fragments
    b0.x = bp[0];  b0.y = bp[NT];
    b1.x = bp[16]; b1.y = bp[NT + 16];
    b2.x = bp[32]; b2.y = bp[NT + 32];
    b3.x = bp[48]; b3.y = bp[NT + 48];
    c0 = __builtin_amdgcn_wmma_f32_16x16x4_f32(false, a, false, b0, (short)0, c0, false, false);
    c1 = __builtin_amdgcn_wmma_f32_16x16x4_f32(false, a, false, b1, (short)0, c1, false, false);
    c2 = __builtin_amdgcn_wmma_f32_16x16x4_f32(false, a, false, b2, (short)0, c2, false, false);
    c3 = __builtin_amdgcn_wmma_f32_16x16x4_f32(false, a, false, b3, (short)0, c3, false, false);
  }

  // D layout: VGPR v, lane L -> (m = v + 8*(L>=16), n = L%16).
  // Output streams 403MB > 192MB L2 -> non-temporal stores.
  float* obase = out + (size_t)(rc0 + 8*half)*NT + p0 + l16;
#pragma unroll
  for (int v = 0; v < 8; ++v) {
    float* orow = obase + (size_t)v*NT;
    __builtin_nontemporal_store(c0[v], orow);
    __builtin_nontemporal_store(c1[v], orow + 16);
    __builtin_nontemporal_store(c2[v], orow + 32);
    __builtin_nontemporal_store(c3[v], orow + 48);
  }
}

// ---------------------------------------------------------------------------
extern "C" void kernel_launch(void* const* d_in, const int* in_sizes, int n_in,
                              void* d_out, int out_size, void* d_ws, size_t ws_size,
                              hipStream_t stream) {
  const float* diff   = (const float*)d_in[0];  // (192,128,3) f32
  const float* weight = (const float*)d_in[1];  // (6144,)     f32
  float* out = (float*)d_out;                   // (64,64,192,128) f32

  // Workspace: Phi (36*24576 f32 = 3.375 MB) then T (4096*36 f32 = 576 KB)
  float* phi = (float*)d_ws;
  float* T   = phi + (size_t)KDIM * NPAIR;

  se3_build_phi<<<(NPAIR + 255)/256, 256, 0, stream>>>(diff, phi);
  se3_build_T  <<<(NRC   + 255)/256, 256, 0, stream>>>(weight, T);
  // 256 row-tiles * 384 pair-tiles = 98304 waves = 12288 blocks of 8 waves
  se3_gemm_wmma<<<12288, 256, 0, stream>>>(T, phi, out);
}